// MultiGraphGCN_11510512354046
// MI455X (gfx1250) — compile-verified
//
#include <hip/hip_runtime.h>

typedef float v2f __attribute__((ext_vector_type(2)));
typedef float v8f __attribute__((ext_vector_type(8)));

#define G_ 4
#define N_ 50000
#define E_ 800000
#define FIN_ 64
#define HID_ 128
#define OUT_ 64

// ---------------- fill / rsqrt ----------------
__global__ void fill_kernel(float* __restrict__ p, float v, long long n) {
  long long i = (long long)blockIdx.x * blockDim.x + threadIdx.x;
  if (i < n) p[i] = v;
}

__global__ void rsqrt_kernel(float* __restrict__ p, long long n) {
  long long i = (long long)blockIdx.x * blockDim.x + threadIdx.x;
  if (i < n) p[i] = rsqrtf(p[i]);
}

// ---------------- degree accumulation ----------------
__global__ void deg_kernel(const int* __restrict__ ei, float* __restrict__ deg,
                           int nEdges, int nNodes) {
  long long i = (long long)blockIdx.x * blockDim.x + threadIdx.x;
  long long total = (long long)G_ * nEdges;
  if (i >= total) return;
  int g = (int)(i / nEdges);
  int e = (int)(i % nEdges);
  int dst = ei[((long long)g * 2 + 1) * nEdges + e];
  atomicAdd(&deg[(long long)g * nNodes + dst], 1.0f);
}

// ---------------- dense GEMM via WMMA f32 16x16x4 ----------------
// H[g, 0:N, 0:FOUT] = X[g, 0:N, 0:K] @ W[g, 0:K, 0:FOUT]
// One wave computes one 16x16 output tile. N divisible by 16 (50000 = 3125*16).
// A 16x4 f32 layout: lanes 0-15 -> M=lane, VGPR{0,1}=K{0,1}; lanes 16-31 -> K{2,3}
// B 4x16 f32 layout: lanes 0-15 -> N=lane, VGPR{0,1}=K{0,1}; lanes 16-31 -> K{2,3}
// C/D 16x16 f32: VGPR j, lanes 0-15 -> M=j, N=lane; lanes 16-31 -> M=j+8, N=lane-16
template <int K, int FOUT>
__global__ void gemm_wmma_kernel(const float* __restrict__ X,
                                 const float* __restrict__ W,
                                 float* __restrict__ H) {
  const int lane = threadIdx.x & 31;
  const int wave = threadIdx.x >> 5;
  const int tilesN = FOUT / 16;
  const long long tilesM = N_ / 16;
  const long long tilesPerGraph = tilesM * tilesN;
  long long tile = (long long)blockIdx.x * (blockDim.x >> 5) + wave;
  if (tile >= (long long)G_ * tilesPerGraph) return;  // wave-uniform exit

  const int g  = (int)(tile / tilesPerGraph);
  const int t  = (int)(tile % tilesPerGraph);
  const int tm = t / tilesN;
  const int tn = t % tilesN;

  const int m   = lane & 15;       // row within A tile / col within B,C,D tile
  const int kh  = lane >> 4;       // K-half select (0: K=0,1  1: K=2,3)
  const int col = tn * 16 + m;

  const float* Xrow = X + ((long long)g * N_ + (long long)tm * 16 + m) * K;
  const float* Wg   = W + (long long)g * K * FOUT;

  v8f acc = {};
#pragma unroll
  for (int kk = 0; kk < K; kk += 4) {
    const int k0 = kk + 2 * kh;
    v2f a = *(const v2f*)(Xrow + k0);               // K consecutive, 8B aligned
    v2f b;
    b.x = Wg[(long long)k0 * FOUT + col];
    b.y = Wg[(long long)(k0 + 1) * FOUT + col];
    acc = __builtin_amdgcn_wmma_f32_16x16x4_f32(
        /*neg_a=*/false, a, /*neg_b=*/false, b,
        /*c_mod=*/(short)0, acc, /*reuse_a=*/false, /*reuse_b=*/false);
  }

  float* Hg = H + ((long long)g * N_ + (long long)tm * 16) * FOUT + col;
#pragma unroll
  for (int j = 0; j < 8; ++j) {
    const int row = j + 8 * kh;
    Hg[(long long)row * FOUT] = acc[j];
  }
}

// ---------------- edge scatter: agg[dst,f] += h[src,f] * dinv[src]*dinv[dst] ----------------
__global__ void scatter_kernel(const float* __restrict__ H, const int* __restrict__ ei,
                               const float* __restrict__ dinv, float* __restrict__ AGG,
                               int nEdges, int F) {
  long long idx = (long long)blockIdx.x * blockDim.x + threadIdx.x;
  long long total = (long long)G_ * nEdges * F;
  if (idx >= total) return;
  const int f = (int)(idx % F);
  const long long t = idx / F;
  const int e = (int)(t % nEdges);
  const int g = (int)(t / nEdges);
  const int src = ei[((long long)g * 2    ) * nEdges + e];
  const int dst = ei[((long long)g * 2 + 1) * nEdges + e];
  const float nrm = dinv[(long long)g * N_ + src] * dinv[(long long)g * N_ + dst];
  const float v = H[((long long)g * N_ + src) * F + f] * nrm;
  atomicAdd(&AGG[((long long)g * N_ + dst) * F + f], v);
}

// ---------------- self-loop + bias + ELU (in place on AGG) ----------------
__global__ void act_kernel(float* __restrict__ AGG, const float* __restrict__ H,
                           const float* __restrict__ dinv, const float* __restrict__ B,
                           int F) {
  long long idx = (long long)blockIdx.x * blockDim.x + threadIdx.x;
  long long total = (long long)G_ * N_ * F;
  if (idx >= total) return;
  const int f = (int)(idx % F);
  const long long node = idx / F;           // = g*N + n
  const int g = (int)(node / N_);
  const float di = dinv[node];
  float x = AGG[idx] + H[idx] * di * di + B[(long long)g * F + f];
  AGG[idx] = x > 0.0f ? x : (__expf(x) - 1.0f);
}

static inline unsigned cdiv_u(long long a, long long b) {
  return (unsigned)((a + b - 1) / b);
}

extern "C" void kernel_launch(void* const* d_in, const int* in_sizes, int n_in,
                              void* d_out, int out_size, void* d_ws, size_t ws_size,
                              hipStream_t stream) {
  const float* x  = (const float*)d_in[0];   // [G, N, FIN]
  const int*   ei = (const int*)  d_in[1];   // [G, 2, E]
  const float* W1 = (const float*)d_in[2];   // [G, FIN, HID]
  const float* b1 = (const float*)d_in[3];   // [G, HID]
  const float* W2 = (const float*)d_in[4];   // [G, HID, OUT]
  const float* b2 = (const float*)d_in[5];   // [G, OUT]
  float* out = (float*)d_out;                // [G*N, OUT]

  float* ws   = (float*)d_ws;
  float* dinv = ws;                                      // G*N
  float* h    = ws + (long long)G_ * N_;                 // G*N*HID (h1, then h2)
  float* agg  = h + (long long)G_ * N_ * HID_;           // G*N*HID

  const long long nGN   = (long long)G_ * N_;
  const long long nGE   = (long long)G_ * E_;
  const long long nH1   = nGN * HID_;
  const long long nH2   = nGN * OUT_;

  // --- symmetric normalization: dinv = rsqrt(1 + indeg) ---
  fill_kernel <<<cdiv_u(nGN, 256), 256, 0, stream>>>(dinv, 1.0f, nGN);
  deg_kernel  <<<cdiv_u(nGE, 256), 256, 0, stream>>>(ei, dinv, E_, N_);
  rsqrt_kernel<<<cdiv_u(nGN, 256), 256, 0, stream>>>(dinv, nGN);

  // --- layer 1: h1 = x@W1 ; agg1 = scatter ; act1 = elu(agg1 + self + b1) ---
  const long long tiles1 = (long long)G_ * (N_ / 16) * (HID_ / 16);
  gemm_wmma_kernel<FIN_, HID_><<<cdiv_u(tiles1, 8), 256, 0, stream>>>(x, W1, h);
  fill_kernel   <<<cdiv_u(nH1, 256), 256, 0, stream>>>(agg, 0.0f, nH1);
  scatter_kernel<<<cdiv_u(nGE * HID_, 256), 256, 0, stream>>>(h, ei, dinv, agg, E_, HID_);
  act_kernel    <<<cdiv_u(nH1, 256), 256, 0, stream>>>(agg, h, dinv, b1, HID_);

  // --- layer 2: h2 = act1@W2 ; scatter into d_out ; final elu in place ---
  const long long tiles2 = (long long)G_ * (N_ / 16) * (OUT_ / 16);
  gemm_wmma_kernel<HID_, OUT_><<<cdiv_u(tiles2, 8), 256, 0, stream>>>(agg, W2, h);
  fill_kernel   <<<cdiv_u(nH2, 256), 256, 0, stream>>>(out, 0.0f, nH2);
  scatter_kernel<<<cdiv_u(nGE * OUT_, 256), 256, 0, stream>>>(h, ei, dinv, out, E_, OUT_);
  act_kernel    <<<cdiv_u(nH2, 256), 256, 0, stream>>>(out, h, dinv, b2, OUT_);
}